// NSLoss_39633958207819
// MI455X (gfx1250) — compile-verified
//
#include <hip/hip_runtime.h>

typedef __attribute__((ext_vector_type(2))) float v2f;
typedef __attribute__((ext_vector_type(8))) float v8f;

#define BATCH 4
#define NPTS  4096
#define TILES (NPTS / 16)        // 256 tiles of 16 points
#define WAVES_PER_BLOCK 8        // 256 threads = 8 wave32
#define PF_DIST 16               // prefetch 16 tiles (4 KB) ahead

// ---------------------------------------------------------------------------
// Zero the scalar output (harness poisons d_out; we accumulate atomically).
// ---------------------------------------------------------------------------
__global__ void zero_out_kernel(float* out) {
    if (threadIdx.x == 0 && blockIdx.x == 0) out[0] = 0.0f;
}

// ---------------------------------------------------------------------------
// Voxelize (trunc((c - offset)/0.05)) and precompute squared norm.
// vox[i] = (vx, vy, vz, vx^2+vy^2+vz^2)
// ---------------------------------------------------------------------------
__global__ void voxel_prep_kernel(const float* __restrict__ in,
                                  float4* __restrict__ vox, int n) {
    int i = blockIdx.x * blockDim.x + threadIdx.x;
    if (i >= n) return;
    float vx = truncf((in[3 * i + 0] - (-3.0f)) / 0.05f);
    float vy = truncf((in[3 * i + 1] - (-3.0f)) / 0.05f);
    float vz = truncf((in[3 * i + 2] - (-1.0f)) / 0.05f);
    vox[i] = make_float4(vx, vy, vz, vx * vx + vy * vy + vz * vz);
}

// ---------------------------------------------------------------------------
// Pre-swizzle the B operand into WMMA lane layout so the hot loop is a bare
// b64 load per lane. For tile t, lane l (column n = l%16):
//   lanes 0-15  (K=0,1): (-2*y.x, -2*y.y)
//   lanes 16-31 (K=2,3): (-2*y.z,  yy)        // K=3 row carries yy_j
// Combined with A = [x, y, z, 1] this gives D = yy_j - 2*x.y with C = 0.
// ---------------------------------------------------------------------------
__global__ void blayout_prep_kernel(const float4* __restrict__ vox,
                                    float2* __restrict__ bcol, int total) {
    int i = blockIdx.x * blockDim.x + threadIdx.x;   // total = BATCH*TILES*32
    if (i >= total) return;
    int lane = i & 31;
    int tile = i >> 5;                               // global tile (batch-contig)
    float4 q = vox[tile * 16 + (lane & 15)];
    bool hi = lane >= 16;
    float2 o;
    o.x = hi ? (-2.0f * q.z) : (-2.0f * q.x);
    o.y = hi ? q.w           : (-2.0f * q.y);
    bcol[i] = o;
}

// ---------------------------------------------------------------------------
// One wave per (batch, 16-row tile): rmin = min over all column tiles of
// (yy_j - 2 x_i.y_j) via V_WMMA_F32_16X16X4_F32 with inline-0 C, then add
// xx_i per row and atomically accumulate the row-min sum * scale.
// ---------------------------------------------------------------------------
__global__ void __launch_bounds__(256)
chamfer_rowmin_kernel(const float4* __restrict__ Xrow,
                      const float2* __restrict__ Bcol,
                      float* __restrict__ out, float scale) {
    const int lane = threadIdx.x & 31;
    const int wid  = threadIdx.x >> 5;
    const int gw   = blockIdx.x * WAVES_PER_BLOCK + wid;   // global wave id
    const int b    = gw >> 8;                              // / TILES
    const int rt   = gw & (TILES - 1);                     // row tile
    const int m    = lane & 15;
    const bool hi  = lane >= 16;

    // A matrix (16x4 f32, 2 VGPRs), loop-invariant:
    // lanes 0-15 -> K=0,1 = (x, y); lanes 16-31 -> K=2,3 = (z, 1).
    float4 p = Xrow[(b * TILES + rt) * 16 + m];
    v2f a;
    a.x = hi ? p.z : p.x;
    a.y = hi ? 1.0f : p.y;
    const float xx = p.w;                                  // added after min

    v8f rmin;
#pragma unroll
    for (int v = 0; v < 8; ++v) rmin[v] = 3.0e38f;
    const v8f czero = {};                                  // -> inline SRC2 = 0

    const float2* btile = Bcol + (size_t)b * TILES * 32 + lane;
    for (int ct = 0; ct < TILES; ++ct) {
        const float2* cur = &btile[(size_t)ct * 32];
        // Speculative prefetch at a constant +4KB offset from the load base:
        // folds into global_prefetch_b8 <base>, off offset:4096 (no extra
        // address math; OOB translation failures are silently dropped).
        __builtin_prefetch((const char*)cur + PF_DIST * 32 * sizeof(float2), 0, 1);

        float2 qb = *cur;                                  // pre-swizzled B lane data
        v2f bb;
        bb.x = qb.x;
        bb.y = qb.y;

        // D = A x B + 0  ==  yy_j - 2 x_i . y_j  for the 16x16 tile
        v8f d = __builtin_amdgcn_wmma_f32_16x16x4_f32(
            /*neg_a=*/false, a, /*neg_b=*/false, bb,
            /*c_mod=*/(short)0, czero, /*reuse_a=*/false, /*reuse_b=*/false);

#pragma unroll
        for (int v = 0; v < 8; ++v) rmin[v] = fminf(rmin[v], d[v]);
    }

    // Column-min within each 16-lane half (lanes 0-15: rows 0-7, 16-31: rows 8-15).
#pragma unroll
    for (int off = 1; off < 16; off <<= 1) {
#pragma unroll
        for (int v = 0; v < 8; ++v)
            rmin[v] = fminf(rmin[v], __shfl_xor(rmin[v], off, 32));
    }

    // Sum of this half's 8 row-mins, then combine halves.
    float s = 0.0f;
#pragma unroll
    for (int v = 0; v < 8; ++v) s += rmin[v];
    s += __shfl(s, lane ^ 16, 32);                         // all 16 rows

    // Sum of xx over the 16 rows (rows live in lanes 0-15 of each half).
    float t = xx;
#pragma unroll
    for (int off = 1; off < 16; off <<= 1) t += __shfl_xor(t, off, 32);

    if (lane == 0) atomicAdd(out, (s + t) * scale);
}

// ---------------------------------------------------------------------------
// Launch: voxel prep, B-layout prep for both clouds, then two row-min passes.
// ---------------------------------------------------------------------------
extern "C" void kernel_launch(void* const* d_in, const int* in_sizes, int n_in,
                              void* d_out, int out_size, void* d_ws, size_t ws_size,
                              hipStream_t stream) {
    const float* preds = (const float*)d_in[0];   // [B, N, 3] f32
    const float* gts   = (const float*)d_in[1];   // [B, N, 3] f32
    float* out = (float*)d_out;

    const int n  = BATCH * NPTS;                  // 16384 points per cloud
    const int nb = BATCH * TILES * 32;            // B-layout elements per cloud

    float4* voxX  = (float4*)d_ws;                // 256 KB
    float4* voxY  = voxX + n;                     // 256 KB
    float2* bcolX = (float2*)(voxY + n);          // 256 KB
    float2* bcolY = bcolX + nb;                   // 256 KB

    zero_out_kernel<<<1, 32, 0, stream>>>(out);
    voxel_prep_kernel<<<(n + 255) / 256, 256, 0, stream>>>(preds, voxX, n);
    voxel_prep_kernel<<<(n + 255) / 256, 256, 0, stream>>>(gts,   voxY, n);
    blayout_prep_kernel<<<(nb + 255) / 256, 256, 0, stream>>>(voxX, bcolX, nb);
    blayout_prep_kernel<<<(nb + 255) / 256, 256, 0, stream>>>(voxY, bcolY, nb);

    const float scale = 1.0f / (float)(BATCH * NPTS);
    dim3 grid(BATCH * TILES / WAVES_PER_BLOCK);   // 128 blocks x 8 waves
    // dist1: nearest gt for each pred point
    chamfer_rowmin_kernel<<<grid, 256, 0, stream>>>(voxX, bcolY, out, scale);
    // dist2: nearest pred for each gt point (same formula, roles swapped)
    chamfer_rowmin_kernel<<<grid, 256, 0, stream>>>(voxY, bcolX, out, scale);
}